// SelfAttention_68478958567940
// MI455X (gfx1250) — compile-verified
//
#include <hip/hip_runtime.h>
#include <cmath>

// ---------------- problem constants (match reference) ----------------
#define DIM       1024
#define NHEADS    16
#define DHEAD     64
#define NSEQ      1605
#define HW        1600
#define PREFIX    (NSEQ - HW)         // 5
#define QKV_COLS  (3 * DIM)           // 3072
#define MTILES    ((NSEQ + 15) / 16)  // 101 (16-row tiles, attention)
#define MBLKS     ((NSEQ + 63) / 64)  // 26  (64-row blocks, GEMMs)

typedef __attribute__((ext_vector_type(2))) float v2f;
typedef __attribute__((ext_vector_type(8))) float v8f;

// f32 WMMA: D(16x16) = A(16x4) x B(4x16) + C.
// A: lane L, vgpr j -> A[L%16][2*(L>>4)+j]
// B: lane L, vgpr j -> B[2*(L>>4)+j][L%16]
// C/D: lane L, vgpr r -> C[r + 8*(L>>4)][L%16]
static __device__ __forceinline__ v8f wmma_f32(v2f a, v2f b, v8f c) {
    return __builtin_amdgcn_wmma_f32_16x16x4_f32(
        /*neg_a=*/false, a, /*neg_b=*/false, b,
        /*c_mod=*/(short)0, c, /*reuse_a=*/false, /*reuse_b=*/false);
}

static __device__ __forceinline__ float rowmax16(float v) {
    v = fmaxf(v, __shfl_xor(v, 1, 32));
    v = fmaxf(v, __shfl_xor(v, 2, 32));
    v = fmaxf(v, __shfl_xor(v, 4, 32));
    v = fmaxf(v, __shfl_xor(v, 8, 32));
    return v;
}
static __device__ __forceinline__ float rowsum16(float v) {
    v += __shfl_xor(v, 1, 32);
    v += __shfl_xor(v, 2, 32);
    v += __shfl_xor(v, 4, 32);
    v += __shfl_xor(v, 8, 32);
    return v;
}

// ---------------------------------------------------------------------
// C[M x N] = X[M x K] @ W[N x K]^T (+ bias[N]).
// One wave per 64x64 output block: 4x4 register-blocked 16x16 WMMA tiles.
// Per k-step: 8 b64 loads feed 16 v_wmma_f32_16x16x4_f32.
// ---------------------------------------------------------------------
__global__ __launch_bounds__(32)
void wmma_gemm_xwt(const float* __restrict__ X, const float* __restrict__ W,
                   const float* __restrict__ bias, float* __restrict__ C,
                   int M, int N, int K) {
    const int bn   = blockIdx.x;          // N/64 blocks (N multiple of 64)
    const int bm   = blockIdx.y;          // ceil(M/64) blocks
    const int lane = threadIdx.x;
    const int l15  = lane & 15;
    const int lhi  = lane >> 4;
    const int koff = lhi * 2;

    const float* __restrict__ xrow[4];
    const float* __restrict__ wrow[4];
    #pragma unroll
    for (int i = 0; i < 4; ++i) {
        const int m = min(bm * 64 + i * 16 + l15, M - 1);   // clamp; OOB rows not stored
        xrow[i] = X + (size_t)m * K + koff;
        const int n = bn * 64 + i * 16 + l15;               // always in range
        wrow[i] = W + (size_t)n * K + koff;
    }

    v8f acc[4][4];
    #pragma unroll
    for (int i = 0; i < 4; ++i)
        #pragma unroll
        for (int j = 0; j < 4; ++j)
            acc[i][j] = v8f{};

    #pragma unroll 2
    for (int k = 0; k < K; k += 4) {
        v2f a[4], b[4];
        #pragma unroll
        for (int i = 0; i < 4; ++i) a[i] = *(const v2f*)(xrow[i] + k);
        #pragma unroll
        for (int j = 0; j < 4; ++j) b[j] = *(const v2f*)(wrow[j] + k);
        #pragma unroll
        for (int i = 0; i < 4; ++i)
            #pragma unroll
            for (int j = 0; j < 4; ++j)
                acc[i][j] = wmma_f32(a[i], b[j], acc[i][j]);
    }

    #pragma unroll
    for (int j = 0; j < 4; ++j) {
        const int n  = bn * 64 + j * 16 + l15;
        const float bv = bias ? bias[n] : 0.0f;
        #pragma unroll
        for (int i = 0; i < 4; ++i) {
            const int mbase = bm * 64 + i * 16 + 8 * lhi;
            #pragma unroll
            for (int r = 0; r < 8; ++r) {
                const int m = mbase + r;
                if (m < M) C[(size_t)m * N + n] = acc[i][j][r] + bv;
            }
        }
    }
}

// ---------------------------------------------------------------------
// RoPE applied in place to q and k (rows PREFIX..NSEQ-1 of qkv).
// Thread handles the (d, d+32) rotation pair for one (seq,head).
// ---------------------------------------------------------------------
__global__ void rope_kernel(float* __restrict__ qkv, const float* __restrict__ rope) {
    const int idx = blockIdx.x * blockDim.x + threadIdx.x;
    const int total = HW * NHEADS * (DHEAD / 2);
    if (idx >= total) return;
    const int d = idx & 31;
    const int h = (idx >> 5) & (NHEADS - 1);
    const int s = idx >> 9;                 // 0..HW-1
    const int seq = s + PREFIX;

    const float* sinp = rope + (size_t)s * DHEAD;
    const float* cosp = rope + (size_t)HW * DHEAD + (size_t)s * DHEAD;
    const float s0 = sinp[d], c0 = cosp[d];
    const float s1 = sinp[d + 32], c1 = cosp[d + 32];

    #pragma unroll
    for (int which = 0; which < 2; ++which) {   // q then k
        float* t = qkv + (size_t)seq * QKV_COLS + which * DIM + h * DHEAD;
        const float t0 = t[d];
        const float t1 = t[d + 32];
        t[d]      = t0 * c0 - t1 * s0;
        t[d + 32] = t1 * c1 + t0 * s1;
    }
}

// ---------------------------------------------------------------------
// Flash attention: one wave per (16-query tile, head). Streams key tiles,
// online softmax, accumulates O = P @ V into four 16x16 f32 tiles.
// ---------------------------------------------------------------------
__global__ __launch_bounds__(32)
void attn_kernel(const float* __restrict__ qkv, float* __restrict__ ctx) {
    __shared__ __align__(16) float pbuf[16 * 16];  // P tile bounce (C-layout -> A-layout)

    const int qt   = blockIdx.x;
    const int h    = blockIdx.y;
    const int lane = threadIdx.x;
    const int l15  = lane & 15;
    const int lhi  = lane >> 4;
    const float scale = 0.125f;  // 1/sqrt(64)

    // Preload the Q tile as 16 A-fragments (covers d = 0..63 in steps of 4).
    const int qrow = min(qt * 16 + l15, NSEQ - 1);
    const float* __restrict__ qptr = qkv + (size_t)qrow * QKV_COLS + h * DHEAD;
    v2f aq[16];
    #pragma unroll
    for (int kk = 0; kk < 16; ++kk)
        aq[kk] = *(const v2f*)(qptr + kk * 4 + lhi * 2);

    v8f o[4] = {v8f{}, v8f{}, v8f{}, v8f{}};
    float rm[8], rl[8];
    #pragma unroll
    for (int r = 0; r < 8; ++r) { rm[r] = -INFINITY; rl[r] = 0.0f; }

    for (int kt = 0; kt < MTILES; ++kt) {
        // ---- S = (Q tile) @ (K tile)^T, 16 wmma over d ----
        const int krow = min(kt * 16 + l15, NSEQ - 1);
        const float* __restrict__ kptr = qkv + (size_t)krow * QKV_COLS + DIM + h * DHEAD;
        v8f s = {};
        #pragma unroll
        for (int kk = 0; kk < 16; ++kk) {
            v2f b = *(const v2f*)(kptr + kk * 4 + lhi * 2);
            s = wmma_f32(aq[kk], b, s);
        }

        // scale + mask invalid keys (column n = l15 of this tile)
        const bool kvalid = (kt * 16 + l15) < NSEQ;
        #pragma unroll
        for (int r = 0; r < 8; ++r)
            s[r] = kvalid ? s[r] * scale : -INFINITY;

        // ---- online softmax over this key tile ----
        float alpha[8];
        float p[8];
        #pragma unroll
        for (int r = 0; r < 8; ++r) {
            const float tilemax = rowmax16(s[r]);
            const float newm    = fmaxf(rm[r], tilemax);
            alpha[r] = __expf(rm[r] - newm);     // exp(-inf)=0 on first tile
            p[r]     = __expf(s[r] - newm);      // masked cols: exp(-inf)=0
            const float rowsum = rowsum16(p[r]);
            rm[r] = newm;
            rl[r] = rl[r] * alpha[r] + rowsum;
        }

        // rescale running O tiles
        #pragma unroll
        for (int dt = 0; dt < 4; ++dt)
            #pragma unroll
            for (int r = 0; r < 8; ++r)
                o[dt][r] *= alpha[r];

        // ---- bounce P through LDS: C-layout -> A-layout ----
        __syncthreads();
        #pragma unroll
        for (int r = 0; r < 8; ++r)
            pbuf[(r + 8 * lhi) * 16 + l15] = p[r];
        __syncthreads();

        // ---- O += P(16x16) @ V(16x64): 4 k-steps x 4 d-tiles ----
        #pragma unroll
        for (int kk = 0; kk < 4; ++kk) {
            const v2f ap = *(const v2f*)(pbuf + l15 * 16 + kk * 4 + lhi * 2);
            const int kr0 = min(kt * 16 + kk * 4 + lhi * 2,     NSEQ - 1);
            const int kr1 = min(kt * 16 + kk * 4 + lhi * 2 + 1, NSEQ - 1);
            const float* __restrict__ v0 = qkv + (size_t)kr0 * QKV_COLS + 2 * DIM + h * DHEAD;
            const float* __restrict__ v1 = qkv + (size_t)kr1 * QKV_COLS + 2 * DIM + h * DHEAD;
            #pragma unroll
            for (int dt = 0; dt < 4; ++dt) {
                v2f b;
                b.x = v0[dt * 16 + l15];
                b.y = v1[dt * 16 + l15];
                o[dt] = wmma_f32(ap, b, o[dt]);
            }
        }
    }

    // ---- normalize and store ctx[m][h*64 + d] ----
    const int mbase = qt * 16 + 8 * lhi;
    #pragma unroll
    for (int dt = 0; dt < 4; ++dt) {
        #pragma unroll
        for (int r = 0; r < 8; ++r) {
            const int m = mbase + r;
            if (m < NSEQ)
                ctx[(size_t)m * DIM + h * DHEAD + dt * 16 + l15] = o[dt][r] / rl[r];
        }
    }
}

// ---------------------------------------------------------------------
extern "C" void kernel_launch(void* const* d_in, const int* in_sizes, int n_in,
                              void* d_out, int out_size, void* d_ws, size_t ws_size,
                              hipStream_t stream) {
    const float* x      = (const float*)d_in[0];
    const float* rope   = (const float*)d_in[1];
    const float* W_qkv  = (const float*)d_in[2];
    const float* W_proj = (const float*)d_in[3];
    const float* b_proj = (const float*)d_in[4];
    float* out = (float*)d_out;

    float* qkv = (float*)d_ws;                              // NSEQ x 3072
    float* ctx = qkv + (size_t)NSEQ * QKV_COLS;             // NSEQ x 1024

    // 1) qkv = x @ W_qkv^T   (64x64 blocks)
    dim3 g1(QKV_COLS / 64, MBLKS);
    wmma_gemm_xwt<<<g1, 32, 0, stream>>>(x, W_qkv, nullptr, qkv, NSEQ, QKV_COLS, DIM);

    // 2) RoPE on q,k rows [PREFIX, NSEQ)
    const int rthreads = HW * NHEADS * (DHEAD / 2);
    rope_kernel<<<(rthreads + 255) / 256, 256, 0, stream>>>(qkv, rope);

    // 3) flash attention -> ctx
    dim3 g2(MTILES, NHEADS);
    attn_kernel<<<g2, 32, 0, stream>>>(qkv, ctx);

    // 4) out = ctx @ W_proj^T + b_proj   (64x64 blocks)
    dim3 g3(DIM / 64, MBLKS);
    wmma_gemm_xwt<<<g3, 32, 0, stream>>>(ctx, W_proj, b_proj, out, NSEQ, DIM, DIM);
}